// IsocortexSubstrate_40192303956592
// MI455X (gfx1250) — compile-verified
//
#include <hip/hip_runtime.h>

typedef __attribute__((ext_vector_type(2))) float v2f;
typedef __attribute__((ext_vector_type(8))) float v8f;

// ---------------------------------------------------------------------------
// Kernel 1: one wave32 computes the three 16-element dot products
//   k_L = dot(B_L, C_L)  for L in {sensory, association, executive}
// with a single exact-f32 v_wmma_f32_16x16x4_f32, then writes the combined
// gate  (k_s>0 && k_a>0 && k_e>0) ? 1.0f : 0.0f  to gate_out[0].
//
// Trick: A[m,k] = B_L[4r+k]*C_L[4r+k] (pre-multiplied in VALU, rows 0-3
// sensory, 4-7 association, 8-11 executive, 12-15 zero) and the WMMA
// B-operand is all-ones, so D[m,n] = sum_k A[m,k] independent of the B
// operand's lane layout. Only the documented A(16x4) and D(16x16) layouts
// are relied upon:
//   A:  lanes 0-15 hold M=lane, VGPR0=K0, VGPR1=K1; lanes 16-31 K2,K3.
//   D:  VGPR r, lanes 0-15 -> D[r, lane]; lanes 16-31 -> D[8+r, lane-16].
// ---------------------------------------------------------------------------
__global__ void ssm_gate_wmma(const float* __restrict__ Bs, const float* __restrict__ Cs,
                              const float* __restrict__ Ba, const float* __restrict__ Ca,
                              const float* __restrict__ Be, const float* __restrict__ Ce,
                              float* __restrict__ gate_out) {
  __shared__ float red[64];
  const int lane  = threadIdx.x;       // 0..31, single wave
  const int m     = lane & 15;         // A-matrix row held by this lane
  const int kbase = (lane >> 4) * 2;   // K=0,1 (lanes 0-15) ; K=2,3 (lanes 16-31)

  // Branch-free operand selection (v_cndmask, EXEC stays all-ones).
  const float* Bp = (m < 4) ? Bs : ((m < 8) ? Ba : Be);
  const float* Cp = (m < 4) ? Cs : ((m < 8) ? Ca : Ce);
  const int   r     = m & 3;
  const int   i0    = 4 * r + kbase;
  const float valid = (m < 12) ? 1.0f : 0.0f;

  v2f a;
  a.x = Bp[i0]     * Cp[i0]     * valid;
  a.y = Bp[i0 + 1] * Cp[i0 + 1] * valid;

  v2f b;
  b.x = 1.0f;  // all-ones B-matrix => D[m,n] = sum_k A[m,k] (layout-invariant)
  b.y = 1.0f;

  v8f c = {};
  c = __builtin_amdgcn_wmma_f32_16x16x4_f32(
      /*neg_a=*/false, a, /*neg_b=*/false, b,
      /*c_mod=*/(short)0, c, /*reuse_a=*/false, /*reuse_b=*/false);

  // lane 0 : lo = D[0..3,0]  = dot_sensory ; hi = D[4..7,0]  = dot_assoc
  // lane 16: lo = D[8..11,0] = dot_executive
  float lo = c[0] + c[1] + c[2] + c[3];
  float hi = c[4] + c[5] + c[6] + c[7];
  red[lane]      = lo;
  red[32 + lane] = hi;
  __syncthreads();
  if (lane == 0) {
    float dot_s = red[0];
    float dot_a = red[32];
    float dot_e = red[16];
    gate_out[0] = (dot_s > 0.0f && dot_a > 0.0f && dot_e > 0.0f) ? 1.0f : 0.0f;
  }
}

// ---------------------------------------------------------------------------
// Kernel 2: bandwidth-bound gated copy.  out[i] = x[i] * g,  g in {0,1}.
// float4 (b128) loads/stores; 4096x1024 f32 = 1,048,576 float4 elements.
// 33.5 MB total traffic -> ~1.4 us at 23.3 TB/s.
// ---------------------------------------------------------------------------
__global__ void gated_stream(const float4* __restrict__ x,
                             const float* __restrict__ gate,
                             float4* __restrict__ out, int n4) {
  const float g = gate[0];  // uniform address -> scalar load, L2 resident
  int i = blockIdx.x * blockDim.x + threadIdx.x;
  if (i < n4) {
    float4 v = x[i];
    out[i] = make_float4(v.x * g, v.y * g, v.z * g, v.w * g);
  }
}

extern "C" void kernel_launch(void* const* d_in, const int* in_sizes, int n_in,
                              void* d_out, int out_size, void* d_ws, size_t ws_size,
                              hipStream_t stream) {
  // setup_inputs order:
  // 0: incoming_spikes (4096*1024 f32)
  // 1: A_sensory   2: B_sensory (16)   3: C_sensory (16)
  // 4: A_assoc     5: B_assoc   (16)   6: C_assoc   (16)
  // 7: A_exec      8: B_exec    (16)   9: C_exec    (16)
  // A_* are mathematically dead (h_prev == 0).
  const float* x  = (const float*)d_in[0];
  const float* Bs = (const float*)d_in[2];
  const float* Cs = (const float*)d_in[3];
  const float* Ba = (const float*)d_in[5];
  const float* Ca = (const float*)d_in[6];
  const float* Be = (const float*)d_in[8];
  const float* Ce = (const float*)d_in[9];

  float* gate = (float*)d_ws;    // ws[0] holds the combined gate
  float* out  = (float*)d_out;

  ssm_gate_wmma<<<1, 32, 0, stream>>>(Bs, Cs, Ba, Ca, Be, Ce, gate);

  const int n4      = out_size / 4;              // 1,048,576 float4s
  const int threads = 256;                       // 8 waves / workgroup
  const int blocks  = (n4 + threads - 1) / threads;
  gated_stream<<<blocks, threads, 0, stream>>>((const float4*)x, gate,
                                               (float4*)out, n4);
}